// Convert2Image_33543694582286
// MI455X (gfx1250) — compile-verified
//
#include <hip/hip_runtime.h>

// out[b,h,w,:] = glo[b, slic[b,h,w]-1, :]
//   glo : [4, 256, 128] f32   (128 KB per batch -> staged in LDS)
//   slic: [4, 512, 512] i32   (values 1..256)
//   out : [4, 512, 512, 128] f32  (512 MiB, store-bandwidth bound)

typedef float f4  __attribute__((ext_vector_type(4)));
typedef int   vi4 __attribute__((ext_vector_type(4)));

// pointer-to-vi4 in specific address spaces (AS1 = global, AS3 = LDS)
typedef __attribute__((address_space(1))) vi4* global_vi4_p;
typedef __attribute__((address_space(3))) vi4* shared_vi4_p;

#define NBATCH 4
#define SEGS   256
#define CHN    128           // floats per segment row (512 bytes)
#define CHN4   (CHN / 4)     // float4s per row = 32 (== wave32 lanes)
#define IMG_PIX (512 * 512)

#define TPB     256          // 8 waves
#define CHUNKS  256          // blocks per batch
#define PIX_PER_BLOCK (IMG_PIX / CHUNKS)       // 1024
#define WAVES_PER_BLOCK (TPB / 32)             // 8
#define PIX_PER_WAVE (PIX_PER_BLOCK / WAVES_PER_BLOCK)  // 128
#define STAGE_ITERS ((SEGS * CHN4) / TPB)      // 8192 float4 / 256 thr = 32

__global__ __launch_bounds__(TPB) void convert2image_gather(
    const float* __restrict__ glo,
    const int*   __restrict__ slic,
    float*       __restrict__ out)
{
    __shared__ float table[SEGS * CHN];   // 131072 bytes of LDS

    const int b     = blockIdx.x / CHUNKS;
    const int chunk = blockIdx.x % CHUNKS;
    const int tid   = threadIdx.x;

    // ---------------- stage per-batch table into LDS ----------------
    const float* gsrc = glo + (size_t)b * (SEGS * CHN);

#if __has_builtin(__builtin_amdgcn_global_load_async_to_lds_b128)
    #pragma unroll
    for (int i = 0; i < STAGE_ITERS; ++i) {
        const int idx = i * TPB + tid;            // float4 index
        __builtin_amdgcn_global_load_async_to_lds_b128(
            (global_vi4_p)(gsrc + (size_t)idx * 4),
            (shared_vi4_p)(table + (size_t)idx * 4),
            /*offset=*/0, /*cpol=*/0);
    }
  #if __has_builtin(__builtin_amdgcn_s_wait_asynccnt)
    __builtin_amdgcn_s_wait_asynccnt(0);
  #else
    asm volatile("s_wait_asynccnt 0" ::: "memory");
  #endif
#else
    {
        const f4* gsrc4 = (const f4*)gsrc;
        f4*       t4w   = (f4*)table;
        #pragma unroll
        for (int i = 0; i < STAGE_ITERS; ++i) {
            const int idx = i * TPB + tid;
            t4w[idx] = gsrc4[idx];                // global_load_b128 + ds_store_b128
        }
    }
#endif
    __syncthreads();

    // ---------------- gather: four pixels per wave per step ----------------
    const int wave = tid >> 5;
    const int lane = tid & 31;

    const int   pixBase = chunk * PIX_PER_BLOCK + wave * PIX_PER_WAVE;
    const int*  sl      = slic + (size_t)b * IMG_PIX;
    f4*         outB    = (f4*)out + (size_t)b * IMG_PIX * CHN4;
    const f4*   t4      = (const f4*)table;

    #pragma unroll 1
    for (int p = pixBase; p < pixBase + PIX_PER_WAVE; p += 4) {
        const int s0 = sl[p + 0] - 1;
        const int s1 = sl[p + 1] - 1;
        const int s2 = sl[p + 2] - 1;
        const int s3 = sl[p + 3] - 1;

        const f4 v0 = t4[s0 * CHN4 + lane];       // ds_load_b128, wave-uniform row
        const f4 v1 = t4[s1 * CHN4 + lane];
        const f4 v2 = t4[s2 * CHN4 + lane];
        const f4 v3 = t4[s3 * CHN4 + lane];

        // coalesced 512B burst per pixel per wave; NT: don't pollute L2
        __builtin_nontemporal_store(v0, outB + (size_t)(p + 0) * CHN4 + lane);
        __builtin_nontemporal_store(v1, outB + (size_t)(p + 1) * CHN4 + lane);
        __builtin_nontemporal_store(v2, outB + (size_t)(p + 2) * CHN4 + lane);
        __builtin_nontemporal_store(v3, outB + (size_t)(p + 3) * CHN4 + lane);
    }
}

extern "C" void kernel_launch(void* const* d_in, const int* in_sizes, int n_in,
                              void* d_out, int out_size, void* d_ws, size_t ws_size,
                              hipStream_t stream) {
    const float* glo  = (const float*)d_in[0];   // [4,256,128] f32
    const int*   slic = (const int*)d_in[1];     // [4,512,512] i32
    float*       out  = (float*)d_out;           // [4,512,512,128] f32

    dim3 grid(NBATCH * CHUNKS);
    dim3 block(TPB);
    convert2image_gather<<<grid, block, 0, stream>>>(glo, slic, out);
}